// CalibrationAwareLoss_79817672228906
// MI455X (gfx1250) — compile-verified
//
#include <hip/hip_runtime.h>
#include <math.h>

// CDNA5 / gfx1250. Fused calibration-aware loss:
//   - 168 MB streamed once (bandwidth bound, ~7.2 us floor @ 23.3 TB/s)
//   - log-sigmoid identity: log p = -log(1+e), log(1-p) = log p - x  (e=exp(-x))
//     -> only v_exp_f32 / v_log_f32 / v_rcp_f32 hardware ops in the hot loop
//   - deterministic reductions via V_WMMA_F32_16X16X4_F32 ones-matrix trick;
//     no float atomics anywhere -> bitwise-stable across graph replays

typedef float v2f __attribute__((ext_vector_type(2)));
typedef float v4f __attribute__((ext_vector_type(4)));
typedef float v8f __attribute__((ext_vector_type(8)));

namespace {

constexpr int      kThreads     = 256;                 // 8 waves (wave32)
constexpr int      kBlocks1     = 512;
constexpr unsigned kNTotal      = 8u * 1024u * 1024u;  // 8,388,608
constexpr unsigned kPerBlock    = kNTotal / kBlocks1;  // 16384 (64 blocks/image)
constexpr unsigned kVecPerBlock = kPerBlock / 4;       // 4096 float4s
constexpr int      kNMahal      = 8 * 4096;            // 32768

// ---------------------------------------------------------------------------
// Half-wave sum via V_WMMA_F32_16X16X4_F32 with B = ones.
// A (16x4 f32, 2 VGPRs/lane): lane L<16 -> row L, K={0,1}; lane L>=16 ->
// row L-16, K={2,3}. With a = {v, 0}:  D[m][n] = v_m + v_{m+16}.
// D: VGPR j holds D[j][lane] (lanes 0-15) / D[j+8][lane-16] (lanes 16-31),
// so the per-lane sum of the 8 D VGPRs is:
//   lanes 0-15 : sum over m=0..7  of (v_m + v_{m+16})
//   lanes 16-31: sum over m=8..15 of (v_m + v_{m+16})
// Wave total = value@lane0 + value@lane16; both halves are written to LDS by
// the caller (no cross-half shuffle needed). Hardware-fixed summation order
// -> deterministic.
// ---------------------------------------------------------------------------
__device__ __forceinline__ float wmma_half_sum(float v) {
  v2f a; a[0] = v;    a[1] = 0.0f;
  v2f b; b[0] = 1.0f; b[1] = 1.0f;
  v8f c = {0.f, 0.f, 0.f, 0.f, 0.f, 0.f, 0.f, 0.f};
  c = __builtin_amdgcn_wmma_f32_16x16x4_f32(
      /*neg_a=*/false, a, /*neg_b=*/false, b,
      /*c_mod=*/(short)0, c, /*reuse_a=*/false, /*reuse_b=*/false);
  return ((c[0] + c[1]) + (c[2] + c[3])) + ((c[4] + c[5]) + (c[6] + c[7]));
}

// ---------------------------------------------------------------------------
// Kernel 1: streaming pass. Block b owns elements [b*16384, (b+1)*16384),
// entirely inside image b/64. Per-thread register histogram (10 bins x
// {cnt, sum_p, sum_y}) + seg/evid accumulators; 32 WMMA wave reductions at
// block end; per-block partials written to workspace.
// ---------------------------------------------------------------------------
__global__ __launch_bounds__(kThreads)
void fused_main(const v4f* __restrict__ pred, const v4f* __restrict__ lab,
                const v4f* __restrict__ tun,  const v4f* __restrict__ aun,
                const v4f* __restrict__ eun,
                float* __restrict__ part,     // [kBlocks1 * 2] : seg, evid
                float* __restrict__ hist) {   // [kBlocks1 * 30]
  __shared__ float red[8 * 32 * 2];           // [wave][quantity][half]
  const int      t    = threadIdx.x;
  const int      lane = t & 31;
  const int      wave = t >> 5;
  const unsigned b    = blockIdx.x;

  float cnt[10], sp[10], sy[10];
#pragma unroll
  for (int j = 0; j < 10; ++j) { cnt[j] = 0.f; sp[j] = 0.f; sy[j] = 0.f; }
  float seg = 0.f, evid = 0.f;

  const size_t base = (size_t)b * kVecPerBlock;
  for (unsigned i = t; i < kVecPerBlock; i += kThreads) {
    v4f x4 = __builtin_nontemporal_load(&pred[base + i]);
    v4f y4 = __builtin_nontemporal_load(&lab[base + i]);
    v4f u4 = __builtin_nontemporal_load(&tun[base + i]);
    v4f a4 = __builtin_nontemporal_load(&aun[base + i]);
    v4f e4 = __builtin_nontemporal_load(&eun[base + i]);
#pragma unroll
    for (int c = 0; c < 4; ++c) {
      float x = x4[c], y = y4[c], u = u4[c];
      float e = __expf(-x);                        // v_exp_f32
      float p = __builtin_amdgcn_rcpf(1.0f + e);   // sigmoid, v_rcp_f32
      // log p = -log(1+e); log(1-p) = log p - x. Clamp at -100 like the
      // reference (only reachable for |x| >~ 88; kept for exactness).
      float logp = fmaxf(-__logf(1.0f + e), -100.0f);   // v_log_f32
      float l1mp = fmaxf(logp - x,          -100.0f);
      float bce  = -(y * logp + (1.0f - y) * l1mp);
      float w    = __builtin_amdgcn_rcpf(u + 1e-6f);    // v_rcp_f32
      seg  += bce * w;
      evid += a4[c] + e4[c];
      // searchsorted(linspace(0,1,11), p, 'right')-1 == floor(p*10); the
      // overflow bin (p==1.0 -> 10) is dropped, matching the reference.
      int bin = (int)(p * 10.0f);
#pragma unroll
      for (int j = 0; j < 10; ++j) {
        float m = (bin == j) ? 1.0f : 0.0f;
        cnt[j] += m;
        sp[j]   = fmaf(m, p, sp[j]);
        sy[j]   = fmaf(m, y, sy[j]);
      }
    }
  }

  // 32 deterministic wave reductions (30 histogram entries + seg + evid)
  float vals[32];
#pragma unroll
  for (int j = 0; j < 10; ++j) {
    vals[j] = cnt[j]; vals[10 + j] = sp[j]; vals[20 + j] = sy[j];
  }
  vals[30] = seg; vals[31] = evid;

  const int  half      = lane >> 4;          // 0: rows 0..7, 1: rows 8..15
  const bool is_writer = (lane & 15) == 0;   // lane 0 or lane 16
#pragma unroll
  for (int k = 0; k < 32; ++k) {
    float s = wmma_half_sum(vals[k]);
    if (is_writer) red[(wave * 32 + k) * 2 + half] = s;
  }
  __syncthreads();
  if (t < 32) {
    float tot = 0.f;
#pragma unroll
    for (int w = 0; w < 8; ++w)
      tot += red[(w * 32 + t) * 2 + 0] + red[(w * 32 + t) * 2 + 1];
    if (t < 30)       hist[(size_t)b * 30 + t] = tot;
    else if (t == 30) part[2 * b + 0] = tot;
    else              part[2 * b + 1] = tot;
  }
}

// ---------------------------------------------------------------------------
// Kernel 2: OOD term. sum over all 32768 of relu(mahal - 2); mean(ood,axis=1)
// then mean over batch collapses to sum/32768 (applied in finalize).
// ---------------------------------------------------------------------------
__global__ __launch_bounds__(kThreads)
void ood_reduce(const float* __restrict__ mahal, float* __restrict__ oodOut) {
  __shared__ float red[8 * 2];
  const int t    = threadIdx.x;
  const int lane = t & 31;
  const int wave = t >> 5;
  float s = 0.f;
  for (int i = t; i < kNMahal; i += kThreads)
    s += fmaxf(mahal[i] - 2.0f, 0.0f);
  float h = wmma_half_sum(s);
  if ((lane & 15) == 0) red[wave * 2 + (lane >> 4)] = h;
  __syncthreads();
  if (t == 0) {
    float r = 0.f;
#pragma unroll
    for (int w = 0; w < 16; ++w) r += red[w];
    oodOut[0] = r;
  }
}

// ---------------------------------------------------------------------------
// Kernel 3: finalize. Deterministic fixed-order sums of the per-block
// partials; ECE-style calibration loss per image; write 4 outputs.
// ---------------------------------------------------------------------------
__global__ __launch_bounds__(kThreads)
void finalize(const float* __restrict__ part, const float* __restrict__ hist,
              const float* __restrict__ ood,  float* __restrict__ out) {
  __shared__ float H[8 * 30];
  __shared__ float redA[8 * 2], redB[8 * 2];
  const int t    = threadIdx.x;
  const int lane = t & 31;
  const int wave = t >> 5;

  // Per-image histogram totals: image im owns blocks [im*64, im*64+64).
  if (t < 240) {
    const int im = t / 30, j = t % 30;
    float s = 0.f;
    const float* bp = hist + (size_t)(im * 64) * 30 + j;
    for (int kb = 0; kb < 64; ++kb) s += bp[(size_t)kb * 30];
    H[t] = s;
  }
  __syncthreads();

  // seg/evid grand totals over 512 block partials (uniform trip count).
  float segp = 0.f, evp = 0.f;
  for (int i = t; i < kBlocks1; i += kThreads) {
    segp += part[2 * i + 0];
    evp  += part[2 * i + 1];
  }
  float h1 = wmma_half_sum(segp);
  float h2 = wmma_half_sum(evp);
  if ((lane & 15) == 0) {
    redA[wave * 2 + (lane >> 4)] = h1;
    redB[wave * 2 + (lane >> 4)] = h2;
  }
  __syncthreads();

  if (t == 0) {
    float segT = 0.f, evT = 0.f;
#pragma unroll
    for (int w = 0; w < 16; ++w) { segT += redA[w]; evT += redB[w]; }

    const float invN = 1.0f / (float)kNTotal;
    float seg_loss = segT * invN;
    float evid     = evT * invN;

    float cal = 0.f;
    for (int im = 0; im < 8; ++im) {
      float err = 0.f, nvalid = 0.f;
      for (int j = 0; j < 10; ++j) {
        float c = H[im * 30 + j];
        if (c > 0.f) {
          float conf = H[im * 30 + 10 + j] / c;
          float acc  = H[im * 30 + 20 + j] / c;
          err += fabsf(conf - acc);
          nvalid += 1.f;
        }
      }
      cal += (nvalid > 0.f) ? (err / nvalid) : 0.f;
    }
    cal *= (1.0f / 8.0f);

    float uncert = ood[0] * (1.0f / (float)kNMahal) + evid;
    float total  = seg_loss + 1.0f * cal + 0.1f * uncert;
    out[0] = total;
    out[1] = seg_loss;
    out[2] = cal;
    out[3] = uncert;
  }
}

}  // namespace

extern "C" void kernel_launch(void* const* d_in, const int* in_sizes, int n_in,
                              void* d_out, int out_size, void* d_ws, size_t ws_size,
                              hipStream_t stream) {
  (void)in_sizes; (void)n_in; (void)out_size; (void)ws_size;
  const v4f*   pred  = (const v4f*)d_in[0];   // pred_masks
  const v4f*   lab   = (const v4f*)d_in[1];   // pseudo_labels
  const v4f*   tun   = (const v4f*)d_in[2];   // total_uncertainty
  const v4f*   aun   = (const v4f*)d_in[3];   // aleatoric_uncertainty
  const v4f*   eun   = (const v4f*)d_in[4];   // epistemic_uncertainty
  const float* mahal = (const float*)d_in[5]; // mahal_distances
  float* out = (float*)d_out;

  // Workspace layout (floats): [0,1024) block partials (seg,evid),
  // [1024, 1024+512*30) block histograms, then 1 float for ood sum. ~64 KB.
  float* ws   = (float*)d_ws;
  float* part = ws;
  float* hist = ws + 2 * kBlocks1;
  float* ood  = ws + 2 * kBlocks1 + kBlocks1 * 30;

  fused_main<<<kBlocks1, kThreads, 0, stream>>>(pred, lab, tun, aun, eun, part, hist);
  ood_reduce<<<1, kThreads, 0, stream>>>(mahal, ood);
  finalize<<<1, kThreads, 0, stream>>>(part, hist, ood, out);
}